// DifferentiableJPEG_28750511079762
// MI455X (gfx1250) — compile-verified
//
#include <hip/hip_runtime.h>
#include <hip/hip_bf16.h>

typedef float v8f __attribute__((ext_vector_type(8)));
typedef float v4f __attribute__((ext_vector_type(4)));
typedef float v2f __attribute__((ext_vector_type(2)));

#define IMG_W 512
#define HW (512 * 512)

// 8x8 DCT-II matrix, row-major [u][x]
__device__ __constant__ float c_D8[64] = {
    0.35355339f,  0.35355339f,  0.35355339f,  0.35355339f,  0.35355339f,  0.35355339f,  0.35355339f,  0.35355339f,
    0.49039264f,  0.41573481f,  0.27778512f,  0.09754516f, -0.09754516f, -0.27778512f, -0.41573481f, -0.49039264f,
    0.46193977f,  0.19134172f, -0.19134172f, -0.46193977f, -0.46193977f, -0.19134172f,  0.19134172f,  0.46193977f,
    0.41573481f, -0.09754516f, -0.49039264f, -0.27778512f,  0.27778512f,  0.49039264f,  0.09754516f, -0.41573481f,
    0.35355339f, -0.35355339f, -0.35355339f,  0.35355339f,  0.35355339f, -0.35355339f, -0.35355339f,  0.35355339f,
    0.27778512f, -0.49039264f,  0.09754516f,  0.41573481f, -0.41573481f, -0.09754516f,  0.49039264f, -0.27778512f,
    0.19134172f, -0.46193977f,  0.46193977f, -0.19134172f, -0.19134172f,  0.46193977f, -0.46193977f,  0.19134172f,
    0.09754516f, -0.27778512f,  0.41573481f, -0.49039264f,  0.49039264f, -0.41573481f,  0.27778512f, -0.09754516f};

// quality=50 => scale=100 => qtable == base table exactly
__device__ __constant__ float c_QLUM[64] = {
    16.f, 11.f, 10.f, 16.f, 24.f,  40.f,  51.f,  61.f,
    12.f, 12.f, 14.f, 19.f, 26.f,  58.f,  60.f,  55.f,
    14.f, 13.f, 16.f, 24.f, 40.f,  57.f,  69.f,  56.f,
    14.f, 17.f, 22.f, 29.f, 51.f,  87.f,  80.f,  62.f,
    18.f, 22.f, 37.f, 56.f, 68.f, 109.f, 103.f,  77.f,
    24.f, 35.f, 55.f, 64.f, 81.f, 104.f, 113.f,  92.f,
    49.f, 64.f, 78.f, 87.f, 103.f, 121.f, 120.f, 101.f,
    72.f, 92.f, 95.f, 98.f, 112.f, 100.f, 103.f,  99.f};

__device__ __constant__ float c_QCHR[64] = {
    17.f, 18.f, 24.f, 47.f, 99.f, 99.f, 99.f, 99.f,
    18.f, 21.f, 26.f, 66.f, 99.f, 99.f, 99.f, 99.f,
    24.f, 26.f, 56.f, 99.f, 99.f, 99.f, 99.f, 99.f,
    47.f, 66.f, 99.f, 99.f, 99.f, 99.f, 99.f, 99.f,
    99.f, 99.f, 99.f, 99.f, 99.f, 99.f, 99.f, 99.f,
    99.f, 99.f, 99.f, 99.f, 99.f, 99.f, 99.f, 99.f,
    99.f, 99.f, 99.f, 99.f, 99.f, 99.f, 99.f, 99.f,
    99.f, 99.f, 99.f, 99.f, 99.f, 99.f, 99.f, 99.f};

// precomputed reciprocals (avoids IEEE f32 division sequences at runtime)
__device__ __constant__ float c_QLUMI[64] = {
    0.06250000f, 0.09090909f, 0.10000000f, 0.06250000f, 0.04166667f, 0.02500000f, 0.01960784f, 0.01639344f,
    0.08333333f, 0.08333333f, 0.07142857f, 0.05263158f, 0.03846154f, 0.01724138f, 0.01666667f, 0.01818182f,
    0.07142857f, 0.07692308f, 0.06250000f, 0.04166667f, 0.02500000f, 0.01754386f, 0.01449275f, 0.01785714f,
    0.07142857f, 0.05882353f, 0.04545455f, 0.03448276f, 0.01960784f, 0.01149425f, 0.01250000f, 0.01612903f,
    0.05555556f, 0.04545455f, 0.02702703f, 0.01785714f, 0.01470588f, 0.00917431f, 0.00970874f, 0.01298701f,
    0.04166667f, 0.02857143f, 0.01818182f, 0.01562500f, 0.01234568f, 0.00961538f, 0.00884956f, 0.01086957f,
    0.02040816f, 0.01562500f, 0.01282051f, 0.01149425f, 0.00970874f, 0.00826446f, 0.00833333f, 0.00990099f,
    0.01388889f, 0.01086957f, 0.01052632f, 0.01020408f, 0.00892857f, 0.01000000f, 0.00970874f, 0.01010101f};

__device__ __constant__ float c_QCHRI[64] = {
    0.05882353f, 0.05555556f, 0.04166667f, 0.02127660f, 0.01010101f, 0.01010101f, 0.01010101f, 0.01010101f,
    0.05555556f, 0.04761905f, 0.03846154f, 0.01515152f, 0.01010101f, 0.01010101f, 0.01010101f, 0.01010101f,
    0.04166667f, 0.03846154f, 0.01785714f, 0.01010101f, 0.01010101f, 0.01010101f, 0.01010101f, 0.01010101f,
    0.02127660f, 0.01515152f, 0.01010101f, 0.01010101f, 0.01010101f, 0.01010101f, 0.01010101f, 0.01010101f,
    0.01010101f, 0.01010101f, 0.01010101f, 0.01010101f, 0.01010101f, 0.01010101f, 0.01010101f, 0.01010101f,
    0.01010101f, 0.01010101f, 0.01010101f, 0.01010101f, 0.01010101f, 0.01010101f, 0.01010101f, 0.01010101f,
    0.01010101f, 0.01010101f, 0.01010101f, 0.01010101f, 0.01010101f, 0.01010101f, 0.01010101f, 0.01010101f,
    0.01010101f, 0.01010101f, 0.01010101f, 0.01010101f, 0.01010101f, 0.01010101f, 0.01010101f, 0.01010101f};

// gfx1250 has V_TANH_F32; use it if the toolchain exposes the builtin.
#if defined(__has_builtin)
#if __has_builtin(__builtin_amdgcn_tanhf)
#define FAST_TANH(x) __builtin_amdgcn_tanhf(x)
#endif
#endif
#ifndef FAST_TANH
#define FAST_TANH(x) tanhf(x)
#endif

// Compiler-only memory fence: LDS from one wave is hardware-in-order, we just
// have to stop the compiler from reordering cross-lane LDS traffic.
#define LDS_FENCE() asm volatile("" ::: "memory")

static __device__ __forceinline__ v8f wmma4(v2f a, v2f b, v8f c) {
  // D = A(16x4 f32) * B(4x16 f32) + C(16x16 f32)
  return __builtin_amdgcn_wmma_f32_16x16x4_f32(false, a, false, b, (short)0, c,
                                               false, false);
}

static __device__ __forceinline__ v8f vzero8() {
  v8f z;
#pragma unroll
  for (int i = 0; i < 8; ++i) z[i] = 0.0f;
  return z;
}

// One wave (32 lanes) processes one 16x16 spatial tile (= 2x2 JPEG 8x8 blocks)
// across all 3 channels. DCT/IDCT run as 16x16x16 f32 WMMA chains with the
// block-diagonal diag(D8,D8) matrix. 8 waves per block.
__global__ __launch_bounds__(256) void jpeg_fused(const float* __restrict__ in,
                                                  float* __restrict__ out) {
  // per wave: 3 channel tiles (16x16, stride 16) + staging (16 x stride17)
  __shared__ float smem[8 * (3 * 256 + 272)];
  const int lane = threadIdx.x & 31;
  const int wid = threadIdx.x >> 5;
  float* tiles = &smem[wid * 1040];
  float* stg = tiles + 768;

  const int tile = blockIdx.x * 8 + wid;  // 32768 tiles total
  const int b = tile >> 10;               // 1024 tiles per image
  const int ti = (tile >> 5) & 31;
  const int tj = tile & 31;
  const float* inb = in + (size_t)b * 3 * HW + (size_t)(ti * 16) * IMG_W + tj * 16;
  float* outb = out + (size_t)b * 3 * HW + (size_t)(ti * 16) * IMG_W + tj * 16;

  // ---- load 16x16 RGB tile (non-temporal b128), forward color xform -> LDS
#pragma unroll
  for (int p = 0; p < 2; ++p) {
    const int f = lane + 32 * p;  // 0..63 float4 slots
    const int row = f >> 2;       // 0..15
    const int c4 = (f & 3) << 2;  // 0,4,8,12
    const v4f r4 = __builtin_nontemporal_load((const v4f*)(inb + (size_t)row * IMG_W + c4));
    const v4f g4 = __builtin_nontemporal_load((const v4f*)(inb + HW + (size_t)row * IMG_W + c4));
    const v4f b4 = __builtin_nontemporal_load((const v4f*)(inb + 2 * HW + (size_t)row * IMG_W + c4));
    v4f y4, cb4, cr4;
#pragma unroll
    for (int k = 0; k < 4; ++k) {
      const float R = r4[k], G = g4[k], B = b4[k];
      y4[k] = 0.299f * R + 0.587f * G + 0.114f * B - 0.5f;  // +OFF-0.5
      cb4[k] = -0.168736f * R - 0.331264f * G + 0.5f * B;
      cr4[k] = 0.5f * R - 0.418688f * G - 0.081312f * B;
    }
    *(v4f*)(tiles + row * 16 + c4) = y4;
    *(v4f*)(tiles + 256 + row * 16 + c4) = cb4;
    *(v4f*)(tiles + 512 + row * 16 + c4) = cr4;
  }
  LDS_FENCE();

  const int m = lane & 15;   // N (column) index / A-row index
  const int hi = lane >> 4;  // half-wave select

  // A operands held in registers: Dfull = diag(D8,D8) and its transpose,
  // laid out per the 16x4-f32 A layout: VGPR v of chunk kc holds
  // K = 4*kc + v + 2*hi, M = lane%16.
  float Af[8], At[8];
#pragma unroll
  for (int i = 0; i < 8; ++i) {
    const int k = 4 * (i >> 1) + (i & 1) + 2 * hi;
    const bool same = (m >> 3) == (k >> 3);
    Af[i] = same ? c_D8[(m & 7) * 8 + (k & 7)] : 0.0f;  // Dfull(m,k)
    At[i] = same ? c_D8[(k & 7) * 8 + (m & 7)] : 0.0f;  // Dfullᵀ(m,k)
  }

  v8f rec[3];
#pragma unroll
  for (int c = 0; c < 3; ++c) {
    const float* tile_c = tiles + c * 256;
    const float* Q = (c == 0) ? c_QLUM : c_QCHR;
    const float* Qi = (c == 0) ? c_QLUMI : c_QCHRI;
    float qv[8], qi[8];
#pragma unroll
    for (int r = 0; r < 8; ++r) {
      qv[r] = Q[(lane & 7) * 8 + r];   // QTAB[c][u=lane%8][v=r] (transposed)
      qi[r] = Qi[(lane & 7) * 8 + r];  // 1/q as compile-time constant
    }

    // ---- Stage 1: T1 = Dfull * Ytile  (B read row-chunked from channel tile)
    v8f t1 = vzero8();
#pragma unroll
    for (int kc = 0; kc < 4; ++kc) {
      v2f a, bv;
      a.x = Af[2 * kc];
      a.y = Af[2 * kc + 1];
      bv.x = tile_c[(4 * kc + 2 * hi) * 16 + m];
      bv.y = tile_c[(4 * kc + 1 + 2 * hi) * 16 + m];
      t1 = wmma4(a, bv, t1);
    }
    LDS_FENCE();
#pragma unroll
    for (int r = 0; r < 8; ++r) stg[(r + 8 * hi) * 17 + m] = t1[r];
    LDS_FENCE();

    // ---- Stage 2: coefᵀ = Dfull * T1ᵀ  (transposed B read from staging)
    v8f cf = vzero8();
#pragma unroll
    for (int kc = 0; kc < 4; ++kc) {
      v2f a, bv;
      a.x = Af[2 * kc];
      a.y = Af[2 * kc + 1];
      bv.x = stg[m * 17 + 4 * kc + 2 * hi];
      bv.y = stg[m * 17 + 4 * kc + 1 + 2 * hi];
      cf = wmma4(a, bv, cf);
    }

    // ---- quantize + differentiable soft-round + dequantize (elementwise)
#pragma unroll
    for (int r = 0; r < 8; ++r) {
      const float d = cf[r] * qi[r];
      const float fr = d - rintf(d);
      cf[r] = (d + 0.5f * FAST_TANH(15.0f * fr)) * qv[r];  // 50*TEMP = 15
    }

    LDS_FENCE();
#pragma unroll
    for (int r = 0; r < 8; ++r) stg[(r + 8 * hi) * 17 + m] = cf[r];
    LDS_FENCE();

    // ---- Stage 3: S = Dfullᵀ * deqᵀ  (direct B read)
    v8f s = vzero8();
#pragma unroll
    for (int kc = 0; kc < 4; ++kc) {
      v2f a, bv;
      a.x = At[2 * kc];
      a.y = At[2 * kc + 1];
      bv.x = stg[(4 * kc + 2 * hi) * 17 + m];
      bv.y = stg[(4 * kc + 1 + 2 * hi) * 17 + m];
      s = wmma4(a, bv, s);
    }
    LDS_FENCE();
#pragma unroll
    for (int r = 0; r < 8; ++r) stg[(r + 8 * hi) * 17 + m] = s[r];
    LDS_FENCE();

    // ---- Stage 4: rec = Dfullᵀ * Sᵀ  (transposed B read) -> keep in regs
    v8f rc = vzero8();
#pragma unroll
    for (int kc = 0; kc < 4; ++kc) {
      v2f a, bv;
      a.x = At[2 * kc];
      a.y = At[2 * kc + 1];
      bv.x = stg[m * 17 + 4 * kc + 2 * hi];
      bv.y = stg[m * 17 + 4 * kc + 1 + 2 * hi];
      rc = wmma4(a, bv, rc);
    }
    rec[c] = rc;
    LDS_FENCE();
  }

  // ---- inverse color transform + clip; restage RGB through LDS so global
  //      stores go out as b128 (C/D layout: row r+8*hi, col m)
#pragma unroll
  for (int r = 0; r < 8; ++r) {
    const float Y = rec[0][r] + 0.5f;  // ycbcr_rec - OFF folds to this
    const float Cb = rec[1][r];
    const float Cr = rec[2][r];
    float R = Y + 1.402f * Cr;
    float G = Y - 0.344136f * Cb - 0.714136f * Cr;
    float B = Y + 1.772f * Cb;
    R = fminf(fmaxf(R, 0.0f), 1.0f);
    G = fminf(fmaxf(G, 0.0f), 1.0f);
    B = fminf(fmaxf(B, 0.0f), 1.0f);
    const int i = r + 8 * hi;
    tiles[i * 16 + m] = R;
    tiles[256 + i * 16 + m] = G;
    tiles[512 + i * 16 + m] = B;
  }
  LDS_FENCE();

#pragma unroll
  for (int p = 0; p < 2; ++p) {
    const int f = lane + 32 * p;
    const int row = f >> 2;
    const int c4 = (f & 3) << 2;
    const v4f R4 = *(const v4f*)(tiles + row * 16 + c4);
    const v4f G4 = *(const v4f*)(tiles + 256 + row * 16 + c4);
    const v4f B4 = *(const v4f*)(tiles + 512 + row * 16 + c4);
    __builtin_nontemporal_store(R4, (v4f*)(outb + (size_t)row * IMG_W + c4));
    __builtin_nontemporal_store(G4, (v4f*)(outb + HW + (size_t)row * IMG_W + c4));
    __builtin_nontemporal_store(B4, (v4f*)(outb + 2 * HW + (size_t)row * IMG_W + c4));
  }
}

extern "C" void kernel_launch(void* const* d_in, const int* in_sizes, int n_in,
                              void* d_out, int out_size, void* d_ws,
                              size_t ws_size, hipStream_t stream) {
  const float* x = (const float*)d_in[0];
  float* out = (float*)d_out;
  // 32 images * (32x32) 16x16-tiles = 32768 tiles; 8 waves (tiles) per block
  jpeg_fused<<<dim3(4096), dim3(256), 0, stream>>>(x, out);
}